// MultiHeadSelfAttention_19928648253974
// MI455X (gfx1250) — compile-verified
//
#include <hip/hip_runtime.h>
#include <hip/hip_bf16.h>

typedef float v2f __attribute__((ext_vector_type(2)));
typedef float v8f __attribute__((ext_vector_type(8)));
typedef unsigned int v4u __attribute__((ext_vector_type(4)));
typedef int v4i __attribute__((ext_vector_type(4)));
typedef int v8i __attribute__((ext_vector_type(8)));

#define WMMA_F32(a, b, c) \
  __builtin_amdgcn_wmma_f32_16x16x4_f32(false, (a), false, (b), (short)0, (c), false, false)

#if __has_builtin(__builtin_amdgcn_tensor_load_to_lds)
#define HAVE_TDM 1
#else
#define HAVE_TDM 0
#endif

__device__ __forceinline__ void wait_tensorcnt0() {
#if __has_builtin(__builtin_amdgcn_s_wait_tensorcnt)
  __builtin_amdgcn_s_wait_tensorcnt(0);
#else
  asm volatile("s_wait_tensorcnt 0x0" ::: "memory");
#endif
}

#if HAVE_TDM
// Issue a 2-D TENSOR_LOAD_TO_LDS: tile1 rows of tile0 fp32 elements, global
// row stride stride0 (elements), optional LDS row padding (pad codes per D#:
// pad_ival k -> pad after 2<<k DWORDs, pad_amt a -> insert a+1 DWORDs).
__device__ __forceinline__ void tdm_load_2d(unsigned lds_off, const void* gptr,
                                            unsigned tile0, unsigned tile1,
                                            unsigned stride0, int pad_en,
                                            unsigned pad_ival, unsigned pad_amt) {
  const unsigned long long ga = (unsigned long long)(uintptr_t)gptr;
  v4u g0;
  g0[0] = 1u;                                  // count=1, user descriptor
  g0[1] = lds_off;                             // lds_addr (bytes)
  g0[2] = (unsigned)ga;                        // global_addr[31:0]
  g0[3] = (unsigned)((ga >> 32) & 0x01ffffffu) | (2u << 30);  // addr[56:32]|type=2
  unsigned w0 = 2u << 16;                      // data_size = 4 bytes
  if (pad_en) w0 |= (1u << 20) | (pad_ival << 22) | (pad_amt << 25);
  v8i g1;
  g1[0] = (int)w0;                             // mask=0 (not in cluster)
  g1[1] = (int)((tile0 & 0xffffu) << 16);      // tensor_dim0[15:0]
  g1[2] = (int)((tile1 & 0xffffu) << 16);      // tensor_dim0 hi=0 | tensor_dim1 lo
  g1[3] = (int)((tile0 & 0xffffu) << 16);      // tensor_dim1 hi=0 | tile_dim0
  g1[4] = (int)(tile1 & 0xffffu);              // tile_dim1 | tile_dim2=0
  g1[5] = (int)stride0;                        // tensor_dim0_stride[31:0]
  g1[6] = 0;
  g1[7] = 0;
  const v4i z4 = {0, 0, 0, 0};
#if __clang_major__ >= 23
  const v8i z8 = {0, 0, 0, 0, 0, 0, 0, 0};
  __builtin_amdgcn_tensor_load_to_lds(g0, g1, z4, z4, z8, 0);
#else
  __builtin_amdgcn_tensor_load_to_lds(g0, g1, z4, z4, 0);
#endif
}
#endif  // HAVE_TDM

// ---------------------------------------------------------------------------
// Kernel 1: LayerNorm over channels.
// x: (B=8, C=512, HW=1024) -> xn: (B*HW, C) row-major.
// ---------------------------------------------------------------------------
__global__ __launch_bounds__(256) void ln512_kernel(
    const float* __restrict__ x, const float* __restrict__ lw,
    const float* __restrict__ lb, float* __restrict__ xn) {
  __shared__ float tile[16][516];
  const int tid = threadIdx.x;
  const int b = blockIdx.x >> 6;
  const int p0 = (blockIdx.x & 63) << 4;
  const int tp = tid & 15;
  const int tc = tid >> 4;
  const float* xb = x + ((size_t)b << 19);

  for (int s = 0; s < 32; ++s) {
    const int c = (s << 4) + tc;
    tile[tp][c] = xb[((size_t)c << 10) + p0 + tp];
  }
  __syncthreads();

  const int g = tid >> 4;
  const int j = tid & 15;
  float sum = 0.0f, sq = 0.0f;
#pragma unroll
  for (int i = 0; i < 32; ++i) {
    const float v = tile[g][j + (i << 4)];
    sum += v; sq += v * v;
  }
#pragma unroll
  for (int d = 1; d < 16; d <<= 1) {
    sum += __shfl_xor(sum, d, 32);
    sq  += __shfl_xor(sq, d, 32);
  }
  const float mu = sum * (1.0f / 512.0f);
  const float var = sq * (1.0f / 512.0f) - mu * mu;
  const float rs = rsqrtf(var + 1e-5f);

  float* orow = xn + ((size_t)(b * 1024 + p0 + g) << 9);
  for (int i = 0; i < 32; ++i) {
    const int c = j + (i << 4);
    orow[c] = (tile[g][c] - mu) * rs * lw[c] + lb[c];
  }
}

// ---------------------------------------------------------------------------
// Kernel 2/4: fp32 WMMA GEMM, D[m][n] = sum_k A[m][k] * Bw[n][k].
// WG tile 64x128, 8 waves (2x4) x (32x32 per wave, 2x2 WMMA tiles).
// K staged in LDS in chunks of 32; tiles fetched by the Tensor Data Mover
// with double buffering (DMA for chunk k+32 overlaps WMMA on chunk k), LDS
// rows padded 32->36 DWORDs by the TDM pad feature (bank-conflict-free).
// EPI==0: QKV scatter epilogue.  EPI==1: proj (+bias+residual, (B,C,HW)).
// ---------------------------------------------------------------------------
template <int EPI>
__global__ __launch_bounds__(256) void gemm512_wmma(
    const float* __restrict__ Amat, const float* __restrict__ Bmat,
    const float* __restrict__ bias, const float* __restrict__ resid,
    float* __restrict__ out0, float* __restrict__ out1,
    float* __restrict__ out2, int K) {
  extern __shared__ char smem[];
  float* a_s = (float*)smem;                      // [2][64*36]
  float* b_s = (float*)(smem + 2 * 9216);         // [2][128*36]

  const int tid = threadIdx.x;
  const int lane = tid & 31;
  const int wid = tid >> 5;
  const int wm = wid >> 2, wn = wid & 3;
  const int col16 = lane & 15;
  const int half = lane >> 4;
  const int koff = half * 2;

  const float* Ab = Amat;
  const float* Bb = Bmat;
  int bz = 0;
  if (EPI == 1) {
    bz = blockIdx.z;
    Bb = Bmat + (size_t)bz * 1024 * 512;
  }
  const int m0 = blockIdx.y * 64;
  const int n0 = blockIdx.x * 128;

  const v8f vzero = {0.f, 0.f, 0.f, 0.f, 0.f, 0.f, 0.f, 0.f};
  v8f acc[2][2];
#pragma unroll
  for (int tm = 0; tm < 2; ++tm)
#pragma unroll
    for (int tn = 0; tn < 2; ++tn) acc[tm][tn] = vzero;

#if HAVE_TDM
  const unsigned lds_base = __builtin_amdgcn_groupstaticsize();
  if (wid == 0) {
    tdm_load_2d(lds_base, Ab + (size_t)m0 * K, 32, 64, (unsigned)K, 1, 4, 3);
    tdm_load_2d(lds_base + 18432u, Bb + (size_t)n0 * K, 32, 128, (unsigned)K, 1, 4, 3);
    wait_tensorcnt0();
  }
  __syncthreads();
  for (int k0 = 0; k0 < K; k0 += 32) {
    const int cur = (k0 >> 5) & 1;
    if (wid == 0 && (k0 + 32) < K) {  // prefetch next chunk via TDM (overlap)
      tdm_load_2d(lds_base + (cur ^ 1) * 9216u,
                  Ab + (size_t)m0 * K + k0 + 32, 32, 64, (unsigned)K, 1, 4, 3);
      tdm_load_2d(lds_base + 18432u + (cur ^ 1) * 18432u,
                  Bb + (size_t)n0 * K + k0 + 32, 32, 128, (unsigned)K, 1, 4, 3);
    }
    const float* ac = a_s + cur * 2304;
    const float* bc = b_s + cur * 4608;
#pragma unroll
    for (int kk = 0; kk < 32; kk += 4) {
      v2f af[2], bf[2];
#pragma unroll
      for (int t = 0; t < 2; ++t) {
        af[t] = *(const v2f*)(&ac[(wm * 32 + t * 16 + col16) * 36 + kk + koff]);
        bf[t] = *(const v2f*)(&bc[(wn * 32 + t * 16 + col16) * 36 + kk + koff]);
      }
#pragma unroll
      for (int tm = 0; tm < 2; ++tm)
#pragma unroll
        for (int tn = 0; tn < 2; ++tn)
          acc[tm][tn] = WMMA_F32(af[tm], bf[tn], acc[tm][tn]);
    }
    if (wid == 0) wait_tensorcnt0();
    __syncthreads();
  }
#else  // fallback: cooperative float4 loads, single buffer
  for (int k0 = 0; k0 < K; k0 += 32) {
    __syncthreads();
#pragma unroll
    for (int it = 0; it < 2; ++it) {
      const int idx = tid + it * 256;
      const int r = idx >> 3, c4 = (idx & 7) << 2;
      *(float4*)(&a_s[r * 36 + c4]) =
          *(const float4*)(Ab + (size_t)(m0 + r) * K + k0 + c4);
    }
#pragma unroll
    for (int it = 0; it < 4; ++it) {
      const int idx = tid + it * 256;
      const int r = idx >> 3, c4 = (idx & 7) << 2;
      *(float4*)(&b_s[r * 36 + c4]) =
          *(const float4*)(Bb + (size_t)(n0 + r) * K + k0 + c4);
    }
    __syncthreads();
#pragma unroll
    for (int kk = 0; kk < 32; kk += 4) {
      v2f af[2], bf[2];
#pragma unroll
      for (int t = 0; t < 2; ++t) {
        af[t] = *(const v2f*)(&a_s[(wm * 32 + t * 16 + col16) * 36 + kk + koff]);
        bf[t] = *(const v2f*)(&b_s[(wn * 32 + t * 16 + col16) * 36 + kk + koff]);
      }
#pragma unroll
      for (int tm = 0; tm < 2; ++tm)
#pragma unroll
        for (int tn = 0; tn < 2; ++tn)
          acc[tm][tn] = WMMA_F32(af[tm], bf[tn], acc[tm][tn]);
    }
  }
#endif

  // Epilogue
#pragma unroll
  for (int tm = 0; tm < 2; ++tm) {
#pragma unroll
    for (int tn = 0; tn < 2; ++tn) {
#pragma unroll
      for (int i = 0; i < 8; ++i) {
        const int mg = m0 + wm * 32 + tm * 16 + i + half * 8;
        const int ng = n0 + wn * 32 + tn * 16 + col16;
        float v = acc[tm][tn][i];
        if (EPI == 0) {
          v += bias[ng];
          const int b = mg >> 10, p = mg & 1023;
          const int three = ng >> 9, h = (ng >> 6) & 7, d = ng & 63;
          float* dst = (three == 0) ? out0 : ((three == 1) ? out1 : out2);
          dst[((size_t)(((b << 3) + h) << 10) + p) * 64 + d] = v;
        } else {
          const size_t idx = ((size_t)(bz * 512 + mg) << 10) + ng;
          out0[idx] = v + bias[mg] + resid[idx];
        }
      }
    }
  }
}

// ---------------------------------------------------------------------------
// Kernel 3: flash attention; one WG per (128 query rows, b*8+h).
// Q fragments in registers for the whole kernel; K/V blocks of 64 keys pulled
// into LDS by the Tensor Data Mover (K rows padded 64->66 by TDM); online
// softmax with 16-lane shuffle reductions; P goes C-layout -> A-layout
// through a per-wave LDS region (in halves).
// ---------------------------------------------------------------------------
__global__ __launch_bounds__(256) void flash64_kernel(
    const float* __restrict__ Q, const float* __restrict__ Kb,
    const float* __restrict__ Vb, float* __restrict__ O) {
  extern __shared__ char smem[];
  float* k_s = (float*)smem;                      // 64 x 66
  float* v_s = (float*)(smem + 4224 * 4);         // 64 x 64
  float* p_s = (float*)(smem + 8320 * 4);         // 8 waves x 16 x 34

  const int tid = threadIdx.x;
  const int lane = tid & 31;
  const int wid = tid >> 5;
  const int col16 = lane & 15;
  const int half = lane >> 4;
  const int koff = half * 2;
  const int bh = blockIdx.y;  // b*8 + h
  const int q0 = blockIdx.x * 128;
  const size_t base = (size_t)bh * 1024 * 64;
  const float* qp = Q + base;
  const float* kp = Kb + base;
  const float* vp = Vb + base;
  float* p_w = p_s + wid * 16 * 34;

  v2f qf[16];
  {
    const int m = q0 + wid * 16 + col16;
#pragma unroll
    for (int kt = 0; kt < 16; ++kt)
      qf[kt] = *(const v2f*)(qp + (size_t)m * 64 + kt * 4 + koff);
  }

  const v8f vzero = {0.f, 0.f, 0.f, 0.f, 0.f, 0.f, 0.f, 0.f};
  v8f o_acc[4];
#pragma unroll
  for (int dt = 0; dt < 4; ++dt) o_acc[dt] = vzero;
  float mrun[8], lrun[8];
#pragma unroll
  for (int i = 0; i < 8; ++i) { mrun[i] = -3.0e38f; lrun[i] = 0.0f; }
  const float scale = 0.125f;  // 1/sqrt(64)

#if HAVE_TDM
  const unsigned lds_base = __builtin_amdgcn_groupstaticsize();
#endif

  for (int j = 0; j < 1024; j += 64) {
    __syncthreads();
#if HAVE_TDM
    if (wid == 0) {
      tdm_load_2d(lds_base, kp + (size_t)j * 64, 64, 64, 64, 1, 5, 1);
      tdm_load_2d(lds_base + 16896u, vp + (size_t)j * 64, 64, 64, 64, 0, 0, 0);
      wait_tensorcnt0();
    }
#else
#pragma unroll
    for (int it = 0; it < 4; ++it) {
      const int idx = tid + it * 256;
      const int r = idx >> 4, c4 = (idx & 15) << 2;
      *(float4*)(&k_s[r * 66 + c4]) =
          *(const float4*)(kp + (size_t)(j + r) * 64 + c4);
      *(float4*)(&v_s[r * 64 + c4]) =
          *(const float4*)(vp + (size_t)(j + r) * 64 + c4);
    }
#endif
    __syncthreads();

    // S = Q K^T for this wave's 16 rows x 64 key columns.
    v8f s[4];
#pragma unroll
    for (int nt = 0; nt < 4; ++nt) {
      v8f a = vzero;
#pragma unroll
      for (int kt = 0; kt < 16; ++kt) {
        const v2f bfr =
            *(const v2f*)(&k_s[(nt * 16 + col16) * 66 + kt * 4 + koff]);
        a = WMMA_F32(qf[kt], bfr, a);
      }
      s[nt] = a;
    }

    // Online softmax (per row i; each lane-half owns rows i / i+8).
#pragma unroll
    for (int i = 0; i < 8; ++i) {
      float mx = -3.0e38f;
#pragma unroll
      for (int nt = 0; nt < 4; ++nt) {
        s[nt][i] *= scale;
        mx = fmaxf(mx, s[nt][i]);
      }
#pragma unroll
      for (int d = 1; d < 16; d <<= 1) mx = fmaxf(mx, __shfl_xor(mx, d, 32));
      const float mnew = fmaxf(mrun[i], mx);
      float rsum = 0.0f;
#pragma unroll
      for (int nt = 0; nt < 4; ++nt) {
        const float e = __expf(s[nt][i] - mnew);
        s[nt][i] = e;
        rsum += e;
      }
#pragma unroll
      for (int d = 1; d < 16; d <<= 1) rsum += __shfl_xor(rsum, d, 32);
      const float corr = __expf(mrun[i] - mnew);
      lrun[i] = lrun[i] * corr + rsum;
      mrun[i] = mnew;
#pragma unroll
      for (int dt = 0; dt < 4; ++dt) o_acc[dt][i] *= corr;
    }

    // O += P * V via per-wave LDS transpose, in two 32-column halves.
#pragma unroll
    for (int hb = 0; hb < 2; ++hb) {
#pragma unroll
      for (int t = 0; t < 2; ++t) {
        const int nt = hb * 2 + t;
#pragma unroll
        for (int i = 0; i < 8; ++i)
          p_w[(i + half * 8) * 34 + t * 16 + col16] = s[nt][i];
      }
#pragma unroll
      for (int kt = 0; kt < 8; ++kt) {
        const v2f af = *(const v2f*)(&p_w[col16 * 34 + kt * 4 + koff]);
        const int kg = hb * 32 + kt * 4 + koff;
#pragma unroll
        for (int dt = 0; dt < 4; ++dt) {
          v2f bfr;
          bfr.x = v_s[kg * 64 + dt * 16 + col16];
          bfr.y = v_s[(kg + 1) * 64 + dt * 16 + col16];
          o_acc[dt] = WMMA_F32(af, bfr, o_acc[dt]);
        }
      }
    }
  }

  // Finalize: divide by row sums, write (B,HW,C) buffer at col h*64+d.
  const int b = bh >> 3, h = bh & 7;
#pragma unroll
  for (int i = 0; i < 8; ++i) {
    const float inv = 1.0f / lrun[i];
    const int qg = q0 + wid * 16 + i + half * 8;
    const size_t rowbase = ((size_t)(b * 1024 + qg) << 9) + h * 64;
#pragma unroll
    for (int dt = 0; dt < 4; ++dt)
      O[rowbase + dt * 16 + col16] = o_acc[dt][i] * inv;
  }
}

// ---------------------------------------------------------------------------
// Launch: LN -> QKV GEMM -> flash attention -> proj GEMM (+bias+residual).
// Workspace: [0,16MB) xn (reused as attention output), then Q, K, V.
// ---------------------------------------------------------------------------
extern "C" void kernel_launch(void* const* d_in, const int* in_sizes, int n_in,
                              void* d_out, int out_size, void* d_ws,
                              size_t ws_size, hipStream_t stream) {
  const float* x      = (const float*)d_in[0];
  const float* ln_w   = (const float*)d_in[1];
  const float* ln_b   = (const float*)d_in[2];
  const float* qkv_w  = (const float*)d_in[3];
  const float* qkv_b  = (const float*)d_in[4];
  const float* proj_w = (const float*)d_in[5];
  const float* proj_b = (const float*)d_in[6];
  float* out = (float*)d_out;

  const size_t SEG = 16777216;  // 8192 * 512 * 4 bytes
  char* ws = (char*)d_ws;
  float* xn = (float*)(ws);            // also reused as attention output
  float* q  = (float*)(ws + SEG);
  float* k  = (float*)(ws + 2 * SEG);
  float* v  = (float*)(ws + 3 * SEG);

  const size_t gemm_lds  = 2 * (9216 + 18432);   // 55296 B (double-buffered)
  const size_t flash_lds = 12672 * 4;            // 50688 B

  ln512_kernel<<<512, 256, 0, stream>>>(x, ln_w, ln_b, xn);
  gemm512_wmma<0><<<dim3(12, 128), 256, gemm_lds, stream>>>(
      xn, qkv_w, qkv_b, nullptr, q, k, v, 512);
  flash64_kernel<<<dim3(8, 64), 256, flash_lds, stream>>>(q, k, v, xn);
  gemm512_wmma<1><<<dim3(8, 8, 8), 256, gemm_lds, stream>>>(
      proj_w, xn, proj_b, x, out, nullptr, nullptr, 512);
}